// ToCoeff_41506563948882
// MI455X (gfx1250) — compile-verified
//
#include <hip/hip_runtime.h>
#include <hip/hip_bf16.h>

// Cubic B-spline prefilter (periodic) on a (2,3,160,160,160) f32 tensor.
// The causal+anticausal IIR with pole p = sqrt(3)-2 is a circular convolution
// with h[n] = sqrt(3)*p^|n|; |p|^24 ~ 2e-14 so the periodic kernel
//   h_per[d] = sqrt(3) * (p^d + p^(N-d)) / (1 - p^N),  N = 160
// is fp32-exact. Each axis pass is therefore a dense 160x160 circulant matrix
// applied along that axis -> batched GEMM (M = 153600 lines, N = K = 160),
// mapped onto V_WMMA_F32_16X16X4_F32 (native f32 WMMA, matches reference fp32).
//
// Memory bound: 3 passes x ~197 MB = ~590 MB -> ~25 us @ 23.3 TB/s; the
// 98 MB tensor fits in the 192 MB L2 so inter-pass traffic stays on-chip.
// H (100 KB) stays hot in WGP$/L2.
//
// v2: pass geometry (Q,P,S) is now compile-time (template) so line-address
// math lowers to magic-multiply/shift instead of runtime divide loops, and
// the contiguous pass's A-fragment pair merges into one global_load_b64.

typedef __attribute__((ext_vector_type(2))) float v2f;
typedef __attribute__((ext_vector_type(8))) float v8f;

#define NAX 160          // filtered-axis length (N and K of the GEMM)
#define NT  10           // 160 / 16 output column tiles per wave
#define KSTEPS 40        // 160 / 4 K-steps

// ---------------------------------------------------------------------------
// Build the periodic prefilter matrix H[i*160 + j] = h_per[(i - j) mod 160].
// Double precision on device; one-time 25600-thread kernel.
// ---------------------------------------------------------------------------
__global__ void build_H(float* __restrict__ H) {
  int idx = blockIdx.x * blockDim.x + threadIdx.x;
  if (idx >= NAX * NAX) return;
  int i = idx / NAX, j = idx % NAX;
  int d = i - j; if (d < 0) d += NAX;
  const double p = -0.26794919243112270647255365849413;  // sqrt(3) - 2
  double pd = 1.0, pnd = 1.0;
  for (int k = 0; k < d; ++k)        pd  *= p;   // p^d
  for (int k = 0; k < NAX - d; ++k)  pnd *= p;   // p^(N-d)
  double pN = pd * pnd;                          // p^N (~1e-92)
  double h = 1.7320508075688772935274463415059 * (pd + pnd) / (1.0 - pN);
  H[idx] = (float)h;
}

// ---------------------------------------------------------------------------
// One filtering pass along one axis, as a WMMA GEMM:  Out[l, n] = sum_k
// In[l, k] * H[n, k].  Line l's base address = (l / Q) * P + (l % Q); the
// filtered-axis element stride is S.  Q/P/S are compile-time constants.
//   innermost axis: Q = 1,     P = 160,     S = 1
//   middle axis:    Q = 160,   P = 25600,   S = 160
//   outer spatial:  Q = 25600, P = 4096000, S = 25600
// Each wave: 16 lines x full 160 outputs. Per K-step: load one 16x4 A frag,
// reuse it across the 10 N-tiles (B frags come from the hot 100 KB H matrix).
//
// WMMA fragment layouts (CDNA5 ISA 7.12.2, f32, wave32):
//   A 16x4 : lane<16 -> M = lane,    VGPR0 = K0, VGPR1 = K1
//            lane>=16 -> M = lane-16, VGPR0 = K2, VGPR1 = K3
//   B 4x16 : mirror (N = lane%16, K pairs {0,1}/{2,3} by lane half)
//   C/D    : VGPR v -> row v (lanes 0-15) / row v+8 (lanes 16-31), N = lane%16
// ---------------------------------------------------------------------------
template <int Q, int P, int S>
__global__ __launch_bounds__(256) void spline_pass(
    const float* __restrict__ in, float* __restrict__ out,
    const float* __restrict__ H, int nTiles)
{
  const int lane  = threadIdx.x & 31;
  const int wave  = threadIdx.x >> 5;
  const int tile  = blockIdx.x * 8 + wave;      // 16-line strip index
  if (tile >= nTiles) return;                   // wave-uniform: EXEC stays all-1s

  const int col   = lane & 15;                  // 0..15
  const int khalf = lane >> 4;                  // 0 or 1

  // A-fragment source line for this lane (divides by constant -> shifts/muls)
  const int lA    = tile * 16 + col;
  const int baseA = (lA / Q) * P + (lA % Q);

  // D-row base addresses (VGPR v -> line tile*16 + khalf*8 + v)
  int baseD[8];
#pragma unroll
  for (int v = 0; v < 8; ++v) {
    int lD = tile * 16 + khalf * 8 + v;
    baseD[v] = (lD / Q) * P + (lD % Q);
  }

  v8f acc[NT];
#pragma unroll
  for (int t = 0; t < NT; ++t) acc[t] = {};

  for (int ks = 0; ks < KSTEPS; ++ks) {
    const int kb = ks * 4;
    // A fragment: lane holds In[lA, kb + 2*khalf + {0,1}] (b64 when S == 1)
    v2f a;
    a.x = in[baseA + (kb + 2 * khalf + 0) * S];
    a.y = in[baseA + (kb + 2 * khalf + 1) * S];
    // gfx1250 prefetch of the next K-step's A data (speculative; OOB dropped)
    __builtin_prefetch(&in[baseA + (kb + 4 + 2 * khalf) * S], 0, 1);
#pragma unroll
    for (int t = 0; t < NT; ++t) {
      // B fragment: B[k,n] = H[n_global, k_global]
      const float* Hrow = H + (t * 16 + col) * NAX + kb + 2 * khalf;
      v2f b;
      b.x = Hrow[0];
      b.y = Hrow[1];
      acc[t] = __builtin_amdgcn_wmma_f32_16x16x4_f32(
          /*neg_a=*/false, a, /*neg_b=*/false, b,
          /*c_mod=*/(short)0, acc[t], /*reuse_a=*/false, /*reuse_b=*/false);
    }
  }

#pragma unroll
  for (int t = 0; t < NT; ++t) {
#pragma unroll
    for (int v = 0; v < 8; ++v) {
      out[baseD[v] + (t * 16 + col) * S] = acc[t][v];
    }
  }
}

// ---------------------------------------------------------------------------
// Launch: build H, then three passes ping-ponging  in -> out -> tmp -> out.
// Workspace layout: H at d_ws[0..25600), tmp at d_ws + 32768 floats
// (needs ws_size >= 128 KB + 98.4 MB).
// ---------------------------------------------------------------------------
extern "C" void kernel_launch(void* const* d_in, const int* in_sizes, int n_in,
                              void* d_out, int out_size, void* d_ws, size_t ws_size,
                              hipStream_t stream) {
  const float* x   = (const float*)d_in[0];
  float*       out = (float*)d_out;
  float*       H   = (float*)d_ws;
  float*       tmp = (float*)d_ws + 32768;      // 128 KB past H

  (void)in_sizes; (void)n_in; (void)out_size; (void)ws_size;

  build_H<<<dim3((NAX * NAX + 255) / 256), dim3(256), 0, stream>>>(H);

  const int lines = 2 * 3 * NAX * NAX;          // 153600 lines per pass
  const int tiles = lines / 16;                 // 9600 wave-tiles
  dim3 grid((tiles + 7) / 8), block(256);       // 8 waves (wave32) per block

  // Pass 1: innermost axis (contiguous)
  spline_pass<1, NAX, 1>
      <<<grid, block, 0, stream>>>(x, out, H, tiles);
  // Pass 2: middle spatial axis (stride 160)
  spline_pass<NAX, NAX * NAX, NAX>
      <<<grid, block, 0, stream>>>(out, tmp, H, tiles);
  // Pass 3: outer spatial axis (stride 25600)
  spline_pass<NAX * NAX, NAX * NAX * NAX, NAX * NAX>
      <<<grid, block, 0, stream>>>(tmp, out, H, tiles);
}